// GCNLSTMTransformer_62749472195275
// MI455X (gfx1250) — compile-verified
//
#include <hip/hip_runtime.h>
#include <hip/hip_bf16.h>
#include <math.h>

typedef float v2f __attribute__((ext_vector_type(2)));
typedef float v8f __attribute__((ext_vector_type(8)));
typedef unsigned int v4u __attribute__((ext_vector_type(4)));
typedef int v4i __attribute__((ext_vector_type(4)));
typedef int v8i __attribute__((ext_vector_type(8)));

#define NB   128            // batch
#define TT   417            // time steps
#define LL   50             // feature dim
#define NN_  (TT*25)        // 10425 graph nodes
#define EE   30000          // edges
#define NG   200            // lstm gates
#define NGP  208            // padded gates
#define KC   100            // lstm K (x|h)
#define KCP  104            // padded K
#define HG   20             // gcn hidden
#define SQ   83             // idx2 cols
#define SS   128            // attention sequence (= N)
#define BBB  85             // attention batch
#define DM   50             // d_model
#define NH   5
#define HD   10
#define ROWS (SS*BBB)       // 10880
#define FFH  2048
#define EPSV 1e-5f

__device__ __forceinline__ float sigf(float x) { return 1.0f / (1.0f + __expf(-x)); }

#if __has_builtin(__builtin_amdgcn_wmma_f32_16x16x4_f32)
__device__ __forceinline__ v8f wmma_f32(v2f a, v2f b, v8f c) {
    return __builtin_amdgcn_wmma_f32_16x16x4_f32(false, a, false, b, (short)0, c, false, false);
}
#else
__device__ __forceinline__ v8f wmma_f32(v2f a, v2f b, v8f c) { return c; }
#endif

// ---- Tensor Data Mover: load a rows x cols f32 tile (row stride in elems) into LDS ----
#if __has_builtin(__builtin_amdgcn_tensor_load_to_lds) && __has_builtin(__builtin_amdgcn_s_wait_tensorcnt)
#define HAVE_TDM 1
__device__ __forceinline__ void tdm_load_2d_f32(unsigned lds_off, const float* gsrc,
                                                unsigned cols, unsigned rows, unsigned stride)
{
    unsigned long long ga = (unsigned long long)(size_t)gsrc;
    v4u g0;
    g0.x = 1u;                                        // count=1, user descriptor
    g0.y = lds_off;                                   // LDS byte address
    g0.z = (unsigned)(ga & 0xffffffffu);              // global_addr[31:0]
    g0.w = (unsigned)((ga >> 32) & 0x1ffffffu)        // global_addr[56:32]
         | (2u << 30);                                // type = 2 ("image")
    v8i g1;
    g1[0] = (int)(2u << 16);                          // data_size = 4 bytes; mask=0
    g1[1] = (int)((cols & 0xffffu) << 16);            // tensor_dim0[15:0]
    g1[2] = (int)(((cols >> 16) & 0xffffu) | ((rows & 0xffffu) << 16)); // td0 hi | td1 lo
    g1[3] = (int)(((rows >> 16) & 0xffffu) | ((cols & 0xffffu) << 16)); // td1 hi | tile_dim0
    g1[4] = (int)(rows & 0xffffu);                    // tile_dim1; tile_dim2 = 0
    g1[5] = (int)stride;                              // tensor_dim0_stride[31:0]
    g1[6] = 0;                                        // stride0 hi | stride1 lo
    g1[7] = 0;
    v4i z4 = { 0, 0, 0, 0 };
#if defined(__clang_major__) && (__clang_major__ >= 23)
    v8i z8 = { 0, 0, 0, 0, 0, 0, 0, 0 };
    __builtin_amdgcn_tensor_load_to_lds(g0, g1, z4, z4, z8, 0);
#else
    __builtin_amdgcn_tensor_load_to_lds(g0, g1, z4, z4, 0);
#endif
    __builtin_amdgcn_s_wait_tensorcnt(0);
}
#endif

// ---------------- utility -----------------
__global__ void zero3_kernel(float* a, int na, float* b, int nb, float* c, int nc) {
    int i = blockIdx.x * blockDim.x + threadIdx.x;
    int st = gridDim.x * blockDim.x;
    for (int k = i; k < na; k += st) a[k] = 0.f;
    for (int k = i; k < nb; k += st) b[k] = 0.f;
    for (int k = i; k < nc; k += st) c[k] = 0.f;
}

__global__ void deg_kernel(const int* __restrict__ ei, float* __restrict__ deg) {
    int e = blockIdx.x * blockDim.x + threadIdx.x;
    if (e < EE) atomicAdd(&deg[ei[EE + e]], 1.0f);
}

__global__ void dinv_kernel(const float* __restrict__ deg, float* __restrict__ dinv) {
    int n = blockIdx.x * blockDim.x + threadIdx.x;
    if (n < NN_) dinv[n] = rsqrtf(deg[n] + 1.0f);
}

// ---------------- LSTM: one persistent workgroup, WMMA gate GEMM ----------------
// gates(128x200) = [x_t | h](128x100) @ [Wih;Whh]^T(100x200) + bias, per step.
__global__ void __launch_bounds__(1024) lstm_kernel(
    const float* __restrict__ xin, float* __restrict__ xout,
    const float* __restrict__ Wih, const float* __restrict__ Whh,
    const float* __restrict__ bih, const float* __restrict__ bhh,
    const float* __restrict__ h0, const float* __restrict__ c0)
{
    extern __shared__ float sm[];
    float* sW = sm;                    // [KCP][NGP]
    float* sA = sW + KCP * NGP;        // [NB][KCP]  (cols 0-49 x_t, 50-99 h, 100-103 zero)
    float* sG = sA + NB * KCP;         // [NB][NGP]
    float* sC = sG + NB * NGP;         // [NB][LL]
    float* sB = sC + NB * LL;          // [NGP]
    const int tid = threadIdx.x;

    for (int i = tid; i < KCP * NGP; i += blockDim.x) {
        int k = i / NGP, n = i % NGP;
        float w = 0.f;
        if (n < NG) {
            if (k < 50)       w = Wih[n * 50 + k];
            else if (k < 100) w = Whh[n * 50 + (k - 50)];
        }
        sW[i] = w;
    }
    for (int n = tid; n < NGP; n += blockDim.x) sB[n] = (n < NG) ? (bih[n] + bhh[n]) : 0.f;
    for (int i = tid; i < NB * LL; i += blockDim.x) {
        int m = i / LL, j = i % LL;
        sA[m * KCP + 50 + j] = h0[i];
        sC[i] = c0[i];
    }
    for (int m = tid; m < NB; m += blockDim.x) {
        sA[m * KCP + 100] = 0.f; sA[m * KCP + 101] = 0.f;
        sA[m * KCP + 102] = 0.f; sA[m * KCP + 103] = 0.f;
    }
    __syncthreads();

    const int lane = tid & 31, wave = tid >> 5;
    const int l15 = lane & 15, lhi = lane >> 4;

    for (int t = 0; t < TT; ++t) {
        for (int i = tid; i < NB * LL; i += blockDim.x) {
            int m = i / LL, j = i % LL;
            sA[m * KCP + j] = xin[(m * TT + t) * LL + j];
        }
        __syncthreads();

        for (int tile = wave; tile < 8 * 13; tile += 32) {
            int mt = tile & 7, nt = tile >> 3;
            int m0 = mt * 16, n0 = nt * 16;
            float bb = sB[n0 + l15];
            v8f acc = { bb, bb, bb, bb, bb, bb, bb, bb };
            const float* Ap = sA + (m0 + l15) * KCP + (lhi << 1);
            const float* Bp = sW + (lhi << 1) * NGP + n0 + l15;
            for (int kk = 0; kk < KCP; kk += 4) {
                v2f a, b;
                a.x = Ap[kk];            a.y = Ap[kk + 1];
                b.x = Bp[kk * NGP];      b.y = Bp[(kk + 1) * NGP];
                acc = wmma_f32(a, b, acc);
            }
            float* Gp = sG + (m0 + (lhi << 3)) * NGP + n0 + l15;
#pragma unroll
            for (int r = 0; r < 8; ++r) Gp[r * NGP] = acc[r];
        }
        __syncthreads();

        for (int i = tid; i < NB * LL; i += blockDim.x) {
            int m = i / LL, j = i % LL;
            float gi = sG[m * NGP + j];
            float gf = sG[m * NGP + 50 + j];
            float gg = sG[m * NGP + 100 + j];
            float go = sG[m * NGP + 150 + j];
            float c  = sigf(gf) * sC[i] + sigf(gi) * tanhf(gg);
            float h  = sigf(go) * tanhf(c);
            sC[i] = c;
            sA[m * KCP + 50 + j] = h;
            xout[(m * TT + t) * LL + j] = h;
        }
        __syncthreads();
    }
}

// ---------------- GCN ----------------
__global__ void gcn1_init_kernel(const float* __restrict__ x1, const float* __restrict__ dinv,
                                 const float* __restrict__ W1, const float* __restrict__ b1,
                                 float* __restrict__ agg)
{
    long total = (long)NB * NN_ * HG;
    long st = (long)gridDim.x * blockDim.x;
    for (long i = (long)blockIdx.x * blockDim.x + threadIdx.x; i < total; i += st) {
        int f = (int)(i % HG);
        long bn = i / HG;
        int n = (int)(bn % NN_);
        int b = (int)(bn / NN_);
        float g0 = x1[(long)b * TT * LL + n * 2 + 0];
        float g1 = x1[(long)b * TT * LL + n * 2 + 1];
        float d = dinv[n];
        agg[i] = d * d * (g0 * W1[f] + g1 * W1[HG + f]) + b1[f];
    }
}

__global__ void gcn1_scatter_kernel(const float* __restrict__ x1, const float* __restrict__ dinv,
                                    const int* __restrict__ ei, const float* __restrict__ W1,
                                    float* __restrict__ agg)
{
    int e = blockIdx.x;
    int r = ei[e], c = ei[EE + e];
    float w = dinv[r] * dinv[c];
    for (int i = threadIdx.x; i < NB * HG; i += blockDim.x) {
        int f = i % HG, b = i / HG;
        float g0 = x1[(long)b * TT * LL + r * 2 + 0];
        float g1 = x1[(long)b * TT * LL + r * 2 + 1];
        float val = w * (g0 * W1[f] + g1 * W1[HG + f]);
        atomicAdd(&agg[((long)b * NN_ + c) * HG + f], val);
    }
}

__global__ void bnstats_kernel(const float* __restrict__ agg, float* __restrict__ stats) {
    __shared__ float s1[HG], s2[HG];
    if (threadIdx.x < HG) { s1[threadIdx.x] = 0.f; s2[threadIdx.x] = 0.f; }
    __syncthreads();
    long total = (long)NB * NN_ * HG;
    long st = (long)gridDim.x * blockDim.x;
    for (long i = (long)blockIdx.x * blockDim.x + threadIdx.x; i < total; i += st) {
        float v = agg[i];
        int f = (int)(i % HG);
        atomicAdd(&s1[f], v);
        atomicAdd(&s2[f], v * v);
    }
    __syncthreads();
    if (threadIdx.x < HG) {
        atomicAdd(&stats[threadIdx.x], s1[threadIdx.x]);
        atomicAdd(&stats[HG + threadIdx.x], s2[threadIdx.x]);
    }
}

__global__ void bnfinal_kernel(float* __restrict__ stats, const float* __restrict__ gamma,
                               const float* __restrict__ beta) {
    int f = threadIdx.x;
    if (f < HG) {
        float cnt = (float)((long)NB * NN_);
        float m = stats[f] / cnt;
        float v = stats[HG + f] / cnt - m * m;
        float sc = rsqrtf(v + EPSV) * gamma[f];
        stats[2 * HG + f] = sc;
        stats[3 * HG + f] = beta[f] - m * sc;
    }
}

__global__ void gcn2_init_kernel(const float* __restrict__ agg, const float* __restrict__ stats,
                                 const float* __restrict__ dinv, const float* __restrict__ W2,
                                 const float* __restrict__ b2, float* __restrict__ xo)
{
    long total = (long)NB * NN_;
    long st = (long)gridDim.x * blockDim.x;
    for (long i = (long)blockIdx.x * blockDim.x + threadIdx.x; i < total; i += st) {
        int n = (int)(i % NN_);
        int b = (int)(i / NN_);
        const float* a = agg + i * HG;
        float o0 = 0.f, o1 = 0.f;
#pragma unroll
        for (int f = 0; f < HG; ++f) {
            float h = fmaxf(a[f] * stats[2 * HG + f] + stats[3 * HG + f], 0.f);
            o0 += h * W2[f * 2 + 0];
            o1 += h * W2[f * 2 + 1];
        }
        float d = dinv[n]; float dd = d * d;
        xo[(long)b * TT * LL + n * 2 + 0] = dd * o0 + b2[0];
        xo[(long)b * TT * LL + n * 2 + 1] = dd * o1 + b2[1];
    }
}

__global__ void gcn2_scatter_kernel(const float* __restrict__ agg, const float* __restrict__ stats,
                                    const float* __restrict__ dinv, const int* __restrict__ ei,
                                    const float* __restrict__ W2, float* __restrict__ xo)
{
    int e = blockIdx.x;
    int r = ei[e], c = ei[EE + e];
    float w = dinv[r] * dinv[c];
    for (int b = threadIdx.x; b < NB; b += blockDim.x) {
        const float* a = agg + ((long)b * NN_ + r) * HG;
        float o0 = 0.f, o1 = 0.f;
#pragma unroll
        for (int f = 0; f < HG; ++f) {
            float h = fmaxf(a[f] * stats[2 * HG + f] + stats[3 * HG + f], 0.f);
            o0 += h * W2[f * 2 + 0];
            o1 += h * W2[f * 2 + 1];
        }
        atomicAdd(&xo[(long)b * TT * LL + c * 2 + 0], w * o0);
        atomicAdd(&xo[(long)b * TT * LL + c * 2 + 1], w * o1);
    }
}

// ---------------- generic batched WMMA GEMM (one wave per 16x16 tile) ----------------
// biasMode: 0 none, 1 per-N column bias, 2 per-M row bias
__global__ void gemm_wmma_kernel(const float* __restrict__ A, const float* __restrict__ B,
                                 const float* __restrict__ bias, float* __restrict__ C,
                                 int M, int N, int K, int lda, int ldb, int ldc,
                                 long strA, long strB, long strC, int biasMode, int doRelu)
{
    const int batch = blockIdx.y;
    const float* Ab = A + (long)batch * strA;
    const float* Bb = B + (long)batch * strB;
    float* Cb = C + (long)batch * strC;
    const int lane = threadIdx.x & 31, wave = threadIdx.x >> 5;
    const int l15 = lane & 15, lhi = lane >> 4;
    const int mt = (M + 15) >> 4, nt = (N + 15) >> 4;
    const int tiles = mt * nt;
    const int wpb = blockDim.x >> 5;
    for (int tile = blockIdx.x * wpb + wave; tile < tiles; tile += gridDim.x * wpb) {
        int tm = tile / nt, tn = tile % nt;
        int m0 = tm * 16, n0 = tn * 16;
        int mrow = m0 + l15;
        int ncol = n0 + l15;
        bool mok = mrow < M, nok = ncol < N;
        // pull the operand panels toward the WGP before the WMMA k-loop
        if (mok) __builtin_prefetch(&Ab[(long)mrow * lda], 0, 2);
        if (nok) __builtin_prefetch(&Bb[ncol], 0, 2);
        v8f acc = { 0.f, 0.f, 0.f, 0.f, 0.f, 0.f, 0.f, 0.f };
        if (biasMode == 1) {
            float bb = nok ? bias[ncol] : 0.f;
#pragma unroll
            for (int r = 0; r < 8; ++r) acc[r] = bb;
        } else if (biasMode == 2) {
#pragma unroll
            for (int r = 0; r < 8; ++r) {
                int mm = m0 + r + (lhi << 3);
                acc[r] = (mm < M) ? bias[mm] : 0.f;
            }
        }
        for (int kk = 0; kk < K; kk += 4) {
            int ka = kk + (lhi << 1);
            v2f a, b;
            a.x = (mok && ka     < K) ? Ab[(long)mrow * lda + ka]       : 0.f;
            a.y = (mok && ka + 1 < K) ? Ab[(long)mrow * lda + ka + 1]   : 0.f;
            b.x = (nok && ka     < K) ? Bb[(long)ka * ldb + ncol]       : 0.f;
            b.y = (nok && ka + 1 < K) ? Bb[(long)(ka + 1) * ldb + ncol] : 0.f;
            acc = wmma_f32(a, b, acc);
        }
#pragma unroll
        for (int r = 0; r < 8; ++r) {
            int mm = m0 + r + (lhi << 3);
            if (mm < M && ncol < N) {
                float v = acc[r];
                if (doRelu) v = fmaxf(v, 0.f);
                Cb[(long)mm * ldc + ncol] = v;
            }
        }
    }
}

// ---------------- attention: one block per (b, head), 2-pass streaming softmax ----------------
__global__ void attn_kernel(const float* __restrict__ qkv, float* __restrict__ obuf) {
    __shared__ float qs[SS * HD], ks[SS * HD], vs[SS * HD];
    int b = blockIdx.x / NH, h = blockIdx.x % NH;
    int base = h * HD;
    for (int i = threadIdx.x; i < SS * HD; i += blockDim.x) {
        int s = i / HD, d = i % HD;
        long off = ((long)s * BBB + b) * 150;
        qs[i] = qkv[off + base + d];
        ks[i] = qkv[off + 50 + base + d];
        vs[i] = qkv[off + 100 + base + d];
    }
    __syncthreads();
    int s = threadIdx.x;
    if (s < SS) {
        const float scale = 0.31622776601683794f; // 1/sqrt(10)
        float q[HD];
#pragma unroll
        for (int d = 0; d < HD; ++d) q[d] = qs[s * HD + d];
        float mx = -1e30f;
        for (int t2 = 0; t2 < SS; ++t2) {
            float dot = 0.f;
#pragma unroll
            for (int d = 0; d < HD; ++d) dot += q[d] * ks[t2 * HD + d];
            mx = fmaxf(mx, dot * scale);
        }
        float sum = 0.f, o[HD];
#pragma unroll
        for (int d = 0; d < HD; ++d) o[d] = 0.f;
        for (int t2 = 0; t2 < SS; ++t2) {
            float dot = 0.f;
#pragma unroll
            for (int d = 0; d < HD; ++d) dot += q[d] * ks[t2 * HD + d];
            float p = __expf(dot * scale - mx);
            sum += p;
#pragma unroll
            for (int d = 0; d < HD; ++d) o[d] += p * vs[t2 * HD + d];
        }
        float inv = 1.0f / sum;
        long off = ((long)s * BBB + b) * DM + base;
#pragma unroll
        for (int d = 0; d < HD; ++d) obuf[off + d] = o[d] * inv;
    }
}

__global__ void addln_kernel(const float* __restrict__ x, const float* __restrict__ r,
                             const float* __restrict__ g, const float* __restrict__ b,
                             float* __restrict__ out)
{
    int row = blockIdx.x * blockDim.x + threadIdx.x;
    if (row >= ROWS) return;
    float v[DM];
    float m = 0.f;
#pragma unroll
    for (int d = 0; d < DM; ++d) { v[d] = x[(long)row * DM + d] + r[(long)row * DM + d]; m += v[d]; }
    m *= (1.0f / DM);
    float var = 0.f;
#pragma unroll
    for (int d = 0; d < DM; ++d) { float dd = v[d] - m; var += dd * dd; }
    var *= (1.0f / DM);
    float rs = rsqrtf(var + EPSV);
#pragma unroll
    for (int d = 0; d < DM; ++d) out[(long)row * DM + d] = (v[d] - m) * rs * g[d] + b[d];
}

// ---------------- fused FFN: relu(x@W1+b1)@W2+b2, 16-row tile, hidden in LDS ----------------
// x tile staged into LDS by the Tensor Data Mover (TENSORcnt), WMMA both phases.
__global__ void ff_kernel(const float* __restrict__ X, const float* __restrict__ W1,
                          const float* __restrict__ b1, const float* __restrict__ W2,
                          const float* __restrict__ b2, float* __restrict__ out)
{
    extern __shared__ float sm[];
    float* ts = sm;             // [16][FFH]
    float* xs = ts + 16 * FFH;  // [16][DM]
    const int r0 = blockIdx.x * 16;
    const int lane = threadIdx.x & 31, wave = threadIdx.x >> 5;
    const int l15 = lane & 15, lhi = lane >> 4;

#if defined(HAVE_TDM)
    if (wave == 0) {
        // DMA the 16x50 activation tile straight into LDS, then drain TENSORcnt.
        tdm_load_2d_f32((unsigned)(size_t)xs, X + (size_t)r0 * DM, DM, 16, DM);
    }
#else
    for (int i = threadIdx.x; i < 16 * DM; i += blockDim.x)
        xs[i] = X[(size_t)r0 * DM + i];
#endif
    __builtin_prefetch(&W1[l15], 0, 2);
    __builtin_prefetch(&W2[l15], 0, 2);
    __syncthreads();

    for (int ntl = wave; ntl < FFH / 16; ntl += 8) {
        int n0 = ntl * 16;
        float bb = b1[n0 + l15];
        v8f acc = { bb, bb, bb, bb, bb, bb, bb, bb };
        for (int kk = 0; kk < 52; kk += 4) {
            int ka = kk + (lhi << 1);
            v2f a, b;
            a.x = (ka     < DM) ? xs[l15 * DM + ka]     : 0.f;
            a.y = (ka + 1 < DM) ? xs[l15 * DM + ka + 1] : 0.f;
            b.x = (ka     < DM) ? W1[(long)ka * FFH + n0 + l15]       : 0.f;
            b.y = (ka + 1 < DM) ? W1[(long)(ka + 1) * FFH + n0 + l15] : 0.f;
            acc = wmma_f32(a, b, acc);
        }
        float* tp = ts + (lhi << 3) * FFH + n0 + l15;
#pragma unroll
        for (int r = 0; r < 8; ++r) tp[r * FFH] = fmaxf(acc[r], 0.f);
    }
    __syncthreads();

    if (wave < 4) {
        int n0 = wave * 16;
        int ncol = n0 + l15;
        bool nok = ncol < DM;
        float bb = nok ? b2[ncol] : 0.f;
        v8f acc = { bb, bb, bb, bb, bb, bb, bb, bb };
        for (int kk = 0; kk < FFH; kk += 4) {
            int ka = kk + (lhi << 1);
            v2f a, b;
            a.x = ts[l15 * FFH + ka]; a.y = ts[l15 * FFH + ka + 1];
            b.x = nok ? W2[(long)ka * DM + ncol]       : 0.f;
            b.y = nok ? W2[(long)(ka + 1) * DM + ncol] : 0.f;
            acc = wmma_f32(a, b, acc);
        }
#pragma unroll
        for (int r = 0; r < 8; ++r) {
            int mm = r + (lhi << 3);
            if (nok) out[(long)(r0 + mm) * DM + ncol] = acc[r];
        }
    }
}

// ---------------- loss1: fused fc + gather + MSE reduction ----------------
__global__ void loss1_kernel(const float* __restrict__ x1, const float* __restrict__ x,
                             const int* __restrict__ idx2, const float* __restrict__ fcW,
                             const float* __restrict__ fcb, float* __restrict__ lsum)
{
    __shared__ float red[256];
    float acc = 0.f;
    int total = NB * SQ;
    for (int i = blockIdx.x * blockDim.x + threadIdx.x; i < total; i += gridDim.x * blockDim.x) {
        int nb = i / SQ;
        int t  = idx2[i];
        const float* row = x1 + ((long)nb * TT + t) * LL;
        const float* tgt = x  + ((long)nb * TT + t + 1) * LL;
        float rr[LL];
#pragma unroll
        for (int k = 0; k < LL; ++k) rr[k] = row[k];
        for (int k2 = 0; k2 < LL; ++k2) {
            float p = fcb[k2];
            for (int k = 0; k < LL; ++k) p += rr[k] * fcW[k * LL + k2];
            float d = p - tgt[k2];
            acc += d * d;
        }
    }
    red[threadIdx.x] = acc;
    __syncthreads();
    for (int st = blockDim.x / 2; st > 0; st >>= 1) {
        if ((int)threadIdx.x < st) red[threadIdx.x] += red[threadIdx.x + st];
        __syncthreads();
    }
    if (threadIdx.x == 0) atomicAdd(lsum, red[0]);
}

// ---------------- final: classifier head + log_softmax + losses ----------------
__global__ void final_kernel(const float* __restrict__ hc, const float* __restrict__ W2,
                             const float* __restrict__ b2, const int* __restrict__ y,
                             const float* __restrict__ lsum, float* __restrict__ out)
{
    __shared__ float red[NB];
    int n = threadIdx.x;
    float lg[6];
    float mx = -1e30f;
#pragma unroll
    for (int j = 0; j < 6; ++j) {
        float a = b2[j];
        for (int k = 0; k < DM; ++k) a += hc[n * DM + k] * W2[k * 6 + j];
        lg[j] = a; mx = fmaxf(mx, a);
    }
    float sum = 0.f;
#pragma unroll
    for (int j = 0; j < 6; ++j) sum += __expf(lg[j] - mx);
    float lse = mx + __logf(sum);
#pragma unroll
    for (int j = 0; j < 6; ++j) out[n * 6 + j] = __expf(lg[j] - lse);
    red[n] = lse - lg[y[n]];
    __syncthreads();
    for (int st = NB / 2; st > 0; st >>= 1) {
        if (n < st) red[n] += red[n + st];
        __syncthreads();
    }
    if (n == 0) {
        out[NB * 6]     = lsum[0] * (1.0f / (float)(NB * SQ * LL));
        out[NB * 6 + 1] = red[0] * (1.0f / (float)NB);
    }
}

// =============================== host orchestration ===============================
#define PRM(i) ((const float*)d_in[6 + (i)])

extern "C" void kernel_launch(void* const* d_in, const int* in_sizes, int n_in,
                              void* d_out, int out_size, void* d_ws, size_t ws_size,
                              hipStream_t stream)
{
    (void)in_sizes; (void)n_in; (void)out_size; (void)ws_size;
    const float* x    = (const float*)d_in[0];
    const int*   y    = (const int*)d_in[1];
    const int*   ei   = (const int*)d_in[2];
    const int*   idx2 = (const int*)d_in[3];
    const float* h0   = (const float*)d_in[4];
    const float* c0   = (const float*)d_in[5];

    float* w = (float*)d_ws;
    size_t o = 0;
    float* DEG   = w + o; o += 10432;
    float* DINV  = w + o; o += 10432;
    float* X1A   = w + o; o += (size_t)NB * TT * LL;      // lstm out
    float* X1B   = w + o; o += (size_t)NB * TT * LL;      // gcn out
    float* AGG   = w + o; o += (size_t)NB * NN_ * HG;     // 26.7M floats
    float* STATS = w + o; o += 128;
    float* LSUM  = w + o; o += 16;
    float* ZA    = w + o; o += (size_t)ROWS * DM;
    float* ZB    = w + o; o += (size_t)ROWS * DM;
    float* QKV   = w + o; o += (size_t)ROWS * 150;
    float* OB    = w + o; o += (size_t)ROWS * DM;
    float* PR    = w + o; o += (size_t)ROWS * DM;
    float* FB    = w + o; o += (size_t)ROWS * DM;
    float* HC    = w + o; o += (size_t)NB * DM;

    // graph degrees -> dinv
    zero3_kernel<<<64, 256, 0, stream>>>(DEG, NN_, LSUM, 16, STATS, 128);
    deg_kernel<<<(EE + 255) / 256, 256, 0, stream>>>(ei, DEG);
    dinv_kernel<<<(NN_ + 255) / 256, 256, 0, stream>>>(DEG, DINV);

    const size_t lstmLds = (size_t)(KCP * NGP + NB * KCP + NB * NGP + NB * LL + NGP) * sizeof(float);
    const float* cur = x;
    for (int i = 0; i < 2; ++i) {
        lstm_kernel<<<1, 1024, lstmLds, stream>>>(cur, X1A,
            PRM(10 * i + 0), PRM(10 * i + 1), PRM(10 * i + 2), PRM(10 * i + 3),
            h0 + (size_t)i * NB * LL, c0 + (size_t)i * NB * LL);
        gcn1_init_kernel<<<2048, 256, 0, stream>>>(X1A, DINV, PRM(10 * i + 4), PRM(10 * i + 5), AGG);
        gcn1_scatter_kernel<<<EE, 256, 0, stream>>>(X1A, DINV, ei, PRM(10 * i + 4), AGG);
        zero3_kernel<<<1, 64, 0, stream>>>(STATS, 2 * HG, STATS, 0, STATS, 0);
        bnstats_kernel<<<2048, 256, 0, stream>>>(AGG, STATS);
        bnfinal_kernel<<<1, 32, 0, stream>>>(STATS, PRM(10 * i + 6), PRM(10 * i + 7));
        gcn2_init_kernel<<<2048, 256, 0, stream>>>(AGG, STATS, DINV, PRM(10 * i + 8), PRM(10 * i + 9), X1B);
        gcn2_scatter_kernel<<<EE, 128, 0, stream>>>(AGG, STATS, DINV, ei, PRM(10 * i + 8), X1B);
        cur = X1B;
    }

    // loss1 on x1 (=X1B) with fc fused
    loss1_kernel<<<256, 256, 0, stream>>>(X1B, x, idx2, PRM(20), PRM(21), LSUM);

    // conv1d (kernel=1 over T): per-sample GEMM  z[n] = conv_W(85x417) @ x1[n](417x50)
    gemm_wmma_kernel<<<dim3(3, 128), 256, 0, stream>>>(
        PRM(22), X1B, PRM(23), ZA, 85, DM, TT, TT, DM, DM,
        0L, (long)TT * LL, (long)85 * DM, /*biasMode=*/2, /*relu=*/0);

    // 3 encoder layers on (S=128, B=85, D=50) flattened to ROWS x 50
    for (int l = 0; l < 3; ++l) {
        const int pb = 24 + 12 * l;
        gemm_wmma_kernel<<<dim3(850, 1), 256, 0, stream>>>(
            ZA, PRM(pb + 0), PRM(pb + 1), QKV, ROWS, 150, DM, DM, 150, 150, 0L, 0L, 0L, 1, 0);
        attn_kernel<<<BBB * NH, 128, 0, stream>>>(QKV, OB);
        gemm_wmma_kernel<<<dim3(340, 1), 256, 0, stream>>>(
            OB, PRM(pb + 2), PRM(pb + 3), PR, ROWS, DM, DM, DM, DM, DM, 0L, 0L, 0L, 1, 0);
        addln_kernel<<<(ROWS + 255) / 256, 256, 0, stream>>>(ZA, PR, PRM(pb + 4), PRM(pb + 5), ZB);
        const size_t ffLds = (size_t)(16 * FFH + 16 * DM) * sizeof(float);
        ff_kernel<<<ROWS / 16, 256, ffLds, stream>>>(ZB, PRM(pb + 6), PRM(pb + 7), PRM(pb + 8), PRM(pb + 9), FB);
        addln_kernel<<<(ROWS + 255) / 256, 256, 0, stream>>>(ZB, FB, PRM(pb + 10), PRM(pb + 11), ZA);
    }

    // classifier: relu(flat(128x4250) @ W1(4250x50) + b1)
    gemm_wmma_kernel<<<dim3(4, 1), 256, 0, stream>>>(
        ZA, PRM(60), PRM(61), HC, NB, DM, BBB * DM, BBB * DM, DM, DM, 0L, 0L, 0L, 1, 1);
    final_kernel<<<1, NB, 0, stream>>>(HC, PRM(62), PRM(63), y, LSUM, (float*)d_out);
}